// ResidualVQ_75359496175838
// MI455X (gfx1250) — compile-verified
//
#include <hip/hip_runtime.h>
#include <hip/hip_bf16.h>

// ResidualVQ on MI455X (gfx1250, wave32).
// Distances via V_WMMA_F32_16X16X4_F32 (exact fp32 => argmin matches reference).
// 137 GFLOP total matmul, ~75MB memory traffic: matrix-pipe bound, fp32 WMMA is cheap enough.

typedef __attribute__((ext_vector_type(2))) float v2f;
typedef __attribute__((ext_vector_type(8))) float v8f;

#define QN      8
#define KCODES  1024
#define DIM     256
#define BN      32768          // B*N tokens
#define TOK_WG  128            // tokens per workgroup (8 waves x 16 rows)
#define RS      260            // padded LDS row stride (floats): bank-conflict-free
#define NT      (KCODES / 16)  // 64 code tiles of 16
#define KK      (DIM / 4)      // 64 k-groups of 4 for wmma f32 16x16x4
#define LOSS_SCALE (1.0f / 8388608.0f)  // 1/(B*N*D)

__global__ __launch_bounds__(256)
void rvq_cnorm_kernel(const float* __restrict__ cb, float* __restrict__ cn) {
    // one wave per code row: |e_k|^2 for all Q*K codes
    int c    = blockIdx.x * 8 + (threadIdx.x >> 5);
    int lane = threadIdx.x & 31;
    const float* p = cb + (long)c * DIM;
    float s = 0.f;
    #pragma unroll
    for (int d = 0; d < DIM / 32; ++d) {
        float v = p[lane + d * 32];
        s += v * v;
    }
    #pragma unroll
    for (int off = 16; off >= 1; off >>= 1)
        s += __shfl_xor(s, off, 32);
    if (lane == 0) cn[c] = s;
}

__global__ __launch_bounds__(256)
void rvq_main_kernel(const float* __restrict__ x,
                     const float* __restrict__ cb,
                     const float* __restrict__ cnorm,
                     float* __restrict__ qout,
                     int*   __restrict__ idx_out,
                     float* __restrict__ loss) {
    __shared__ float r_lds[TOK_WG * RS];   // residual, padded rows
    __shared__ float c_lds[16 * RS];       // current 16-code tile
    __shared__ float rn_lds[TOK_WG];       // |r|^2 per token
    __shared__ int   idx_lds[TOK_WG];
    __shared__ float dmin_lds[TOK_WG];
    __shared__ float lsum;

    const int tid   = threadIdx.x;
    const int lane  = tid & 31;
    const int wave  = tid >> 5;
    const int tok0  = blockIdx.x * TOK_WG;
    const int m     = lane & 15;     // column index within 16-lane half (N for B/C, M for A)
    const int khalf = lane >> 4;     // 0/1: which pair of K values this half-lane holds

    // ---- load x -> residual (coalesced float4) ----
    for (int i = tid; i < TOK_WG * (DIM / 4); i += 256) {
        int t = i >> 6, d0 = (i & 63) << 2;
        float4 v = *(const float4*)(x + (tok0 + t) * DIM + d0);
        *(float4*)(r_lds + t * RS + d0) = v;
    }
    __syncthreads();

    for (int q = 0; q < QN; ++q) {
        const int cbq = q * KCODES * DIM;

        // |r|^2 per token (vectorized, rows 16B-aligned since RS*4 % 16 == 0)
        if (tid < TOK_WG) {
            const float4* r4 = (const float4*)(r_lds + tid * RS);
            float s = 0.f;
            #pragma unroll 8
            for (int d = 0; d < DIM / 4; ++d) {
                float4 v = r4[d];
                s += v.x * v.x + v.y * v.y + v.z * v.z + v.w * v.w;
            }
            rn_lds[tid] = s;
        }
        if (tid == 0) lsum = 0.f;

        // ---- hoist A fragments for this wave's 16 token rows (held across all 64 N tiles) ----
        // A layout (f32 16x4): lanes 0-15 -> K = k0,k0+1 ; lanes 16-31 -> K = k0+2,k0+3
        v2f afrag[KK];
        {
            const float* ap = r_lds + (wave * 16 + m) * RS + 2 * khalf;
            #pragma unroll
            for (int kk = 0; kk < KK; ++kk) {
                v2f a;
                a.x = ap[kk * 4 + 0];
                a.y = ap[kk * 4 + 1];
                afrag[kk] = a;
            }
        }

        // running argmin over codes this lane observes (codes with n%16 == m)
        float bestv[8];
        int   besti[8];
        #pragma unroll
        for (int j = 0; j < 8; ++j) { bestv[j] = 3.4e38f; besti[j] = 0; }

        for (int nt = 0; nt < NT; ++nt) {
            // stage 16-code tile into LDS (shared by all 8 waves)
            __syncthreads();
            for (int i = tid; i < 16 * (DIM / 4); i += 256) {
                int n = i >> 6, d0 = (i & 63) << 2;
                float4 v = *(const float4*)(cb + cbq + (nt * 16 + n) * DIM + d0);
                *(float4*)(c_lds + n * RS + d0) = v;
            }
            __syncthreads();

            // acc(16x16) = A(16x256) * B(256x16) via 64 wmma f32 16x16x4
            v8f acc = {};
            const float* bp = c_lds + m * RS + 2 * khalf;
            #pragma unroll
            for (int kk = 0; kk < KK; ++kk) {
                v2f b;
                b.x = bp[kk * 4 + 0];
                b.y = bp[kk * 4 + 1];
                acc = __builtin_amdgcn_wmma_f32_16x16x4_f32(
                    false, afrag[kk], false, b, (short)0, acc, false, false);
            }

            // candidate distance (minus |r|^2, constant in n): |e|^2 - 2 r.e
            float cn   = cnorm[q * KCODES + nt * 16 + m];
            int   nidx = nt * 16 + m;
            #pragma unroll
            for (int j = 0; j < 8; ++j) {
                float dc = __builtin_fmaf(-2.f, acc[j], cn);
                if (dc < bestv[j]) { bestv[j] = dc; besti[j] = nidx; }
            }
        }

        // cross-lane argmin within each 16-lane half (first-min tie-break: lower index)
        #pragma unroll
        for (int j = 0; j < 8; ++j) {
            float v  = bestv[j];
            int   ix = besti[j];
            #pragma unroll
            for (int off = 8; off >= 1; off >>= 1) {
                float ov = __shfl_xor(v, off, 16);
                int   oi = __shfl_xor(ix, off, 16);
                if (ov < v || (ov == v && oi < ix)) { v = ov; ix = oi; }
            }
            if (m == 0) {
                int t = wave * 16 + j + khalf * 8;  // C layout: vgpr j -> M=j (lanes<16), M=j+8 (lanes>=16)
                idx_lds[t]  = ix;
                dmin_lds[t] = v;
            }
        }
        __syncthreads();

        // loss accumulation + index output
        if (tid < TOK_WG) {
            float dm = rn_lds[tid] + dmin_lds[tid];  // = |r - e_idx|^2
            atomicAdd(&lsum, dm);
            idx_out[q * BN + tok0 + tid] = idx_lds[tid];
        }
        __syncthreads();
        if (tid == 0) atomicAdd(loss + q, lsum * LOSS_SCALE);

        // residual -= selected code (coalesced: 64 consecutive threads share one code row)
        for (int i = tid; i < TOK_WG * (DIM / 4); i += 256) {
            int t = i >> 6, d0 = (i & 63) << 2;
            int ix = idx_lds[t];
            float4 c = *(const float4*)(cb + cbq + ix * DIM + d0);
            float4* rp = (float4*)(r_lds + t * RS + d0);
            float4 r = *rp;
            r.x -= c.x; r.y -= c.y; r.z -= c.z; r.w -= c.w;
            *rp = r;
        }
        __syncthreads();
    }

    // quantized_out = x - final_residual
    for (int i = tid; i < TOK_WG * (DIM / 4); i += 256) {
        int t = i >> 6, d0 = (i & 63) << 2;
        float4 xv = *(const float4*)(x + (tok0 + t) * DIM + d0);
        float4 rv = *(const float4*)(r_lds + t * RS + d0);
        float4 o;
        o.x = xv.x - rv.x; o.y = xv.y - rv.y; o.z = xv.z - rv.z; o.w = xv.w - rv.w;
        *(float4*)(qout + (tok0 + t) * DIM + d0) = o;
    }
}

extern "C" void kernel_launch(void* const* d_in, const int* in_sizes, int n_in,
                              void* d_out, int out_size, void* d_ws, size_t ws_size,
                              hipStream_t stream) {
    const float* x  = (const float*)d_in[0];   // [16,2048,256] f32
    const float* cb = (const float*)d_in[1];   // [8,1024,256]  f32

    float* qout    = (float*)d_out;                       // [16,2048,256]
    int*   idx_out = (int*)(qout + (size_t)BN * DIM);     // [8,16,2048] int32
    float* loss    = (float*)(idx_out + (size_t)QN * BN); // [8] f32

    float* cn = (float*)d_ws;  // 8192 floats of code norms

    hipMemsetAsync(loss, 0, QN * sizeof(float), stream);
    rvq_cnorm_kernel<<<(QN * KCODES) / 8, 256, 0, stream>>>(cb, cn);
    rvq_main_kernel<<<BN / TOK_WG, 256, 0, stream>>>(x, cb, cn, qout, idx_out, loss);
}